// MLP_41609643164201
// MI455X (gfx1250) — compile-verified
//
#include <hip/hip_runtime.h>

typedef float v2f __attribute__((ext_vector_type(2)));
typedef float v8f __attribute__((ext_vector_type(8)));

#if defined(__HIP_DEVICE_COMPILE__) && __has_builtin(__builtin_amdgcn_tanhf)
  #define TANHF(x) __builtin_amdgcn_tanhf(x)
#else
  #define TANHF(x) tanhf(x)
#endif

// Swap the hi 16-lane half of `a` with the lo 16-lane half of `b`:
//   a' = {a.lo, b.lo}   b' = {a.hi, b.hi}
// v_permlane16_swap_b32 confirmed live on gfx1250 (round 2 histogram/asm).
__device__ __forceinline__ void half_swap(float& a, float& b, bool loH) {
#if defined(__HIP_DEVICE_COMPILE__) && __has_builtin(__builtin_amdgcn_permlane16_swap)
  auto r = __builtin_amdgcn_permlane16_swap(__float_as_uint(a), __float_as_uint(b),
                                            false, false);
  a = __uint_as_float(r[0]);
  b = __uint_as_float(r[1]);
#else
  const float sa = __shfl_xor(a, 16, 32);
  const float sb = __shfl_xor(b, 16, 32);
  const float na = loH ? a : sb;
  const float nb = loH ? sa : b;
  a = na; b = nb;
#endif
}

struct MlpParams {
  const float* x1; const float* x2;
  const float* w1; const float* w2; const float* b1;
  const float* W3; const float* B3; const float* W4; const float* B4;
  const float* W5; const float* B5; const float* W6; const float* B6;
  const float* W7; const float* B7; const float* W8; const float* B8;
  const float* alpha; const float* w9; const float* b9;
  float* out; int n;
};

// D/C layout (16x16 f32): VGPR r -> lanes 0-15: feature r, sample=lane;
//                                   lanes 16-31: feature r+8, sample=lane-16.
// A (16x4 f32, = alpha*W^T, zero padded): lane m=lane&15 holds row m;
//   VGPR0 = K=(0|2 by half), VGPR1 = K=(1|3 by half).
// B (4x16 f32): VGPR0 = {K=0 lo-half, K=2 hi-half}, VGPR1 = {K=1, K=3}.
// Invariant: padded features 10..15 of every activation D are exactly 0.
__global__ __launch_bounds__(256) void mlp_wmma_kernel(MlpParams p) {
  __shared__ float Wp[6][16][16];           // alpha * W, zero padded 10x10 -> 16x16
  __shared__ float Bp[6][16];               // alpha * bias, zero padded
  __shared__ float w1p[16], w2p[16], b1p[16], w9p[16];

  const int tid = threadIdx.x;
  const float alpha = p.alpha[0];
  const float* Wg[6] = {p.W3, p.W4, p.W5, p.W6, p.W7, p.W8};
  const float* Bg[6] = {p.B3, p.B4, p.B5, p.B6, p.B7, p.B8};

  for (int idx = tid; idx < 6 * 256; idx += 256) {
    const int L = idx >> 8, pos = idx & 255, k = pos >> 4, j = pos & 15;
    Wp[L][k][j] = (k < 10 && j < 10) ? alpha * Wg[L][k * 10 + j] : 0.0f;
  }
  if (tid < 96) {
    const int L = tid >> 4, f = tid & 15;
    Bp[L][f] = (f < 10) ? alpha * Bg[L][f] : 0.0f;
  }
  if (tid < 16) {
    w1p[tid] = (tid < 10) ? alpha * p.w1[tid] : 0.0f;
    w2p[tid] = (tid < 10) ? alpha * p.w2[tid] : 0.0f;
    b1p[tid] = (tid < 10) ? alpha * p.b1[tid] : 0.0f;
    w9p[tid] = (tid < 10) ? p.w9[tid] : 0.0f;       // final layer: no alpha, no tanh
  }
  __syncthreads();

  const int lane = tid & 31;
  const int m    = lane & 15;          // sample-in-tile / A row
  const bool loH = (lane < 16);
  const int koff = loH ? 0 : 2;        // A K offset per lane half
  const int fo   = loH ? 0 : 8;        // feature offset per lane half (D layout)

  // Hoist all weights into registers (per-wave constants).
  v2f A[6][3];
#pragma unroll
  for (int L = 0; L < 6; ++L)
#pragma unroll
    for (int w = 0; w < 3; ++w) {
      A[L][w].x = Wp[L][4 * w + koff + 0][m];
      A[L][w].y = Wp[L][4 * w + koff + 1][m];
    }
  // Per-layer bias in D/C layout, used directly as the WMMA C-matrix init.
  v8f abv[6];
#pragma unroll
  for (int L = 0; L < 6; ++L)
#pragma unroll
    for (int r = 0; r < 8; ++r) abv[L][r] = Bp[L][r + fo];
  float w1s[8], w2s[8], b1s[8], w9s[8];
#pragma unroll
  for (int r = 0; r < 8; ++r) {
    w1s[r] = w1p[r + fo]; w2s[r] = w2p[r + fo];
    b1s[r] = b1p[r + fo]; w9s[r] = w9p[r + fo];
  }
  const float b9v = p.b9[0];

  const int numTiles   = (p.n + 15) >> 4;                      // 16 samples per wave tile
  const int waveId     = blockIdx.x * (blockDim.x >> 5) + (tid >> 5);
  const int waveStride = gridDim.x * (blockDim.x >> 5);

  for (int t = waveId; t < numTiles; t += waveStride) {
    const int base = t << 4;
    int gi = base + m; if (gi >= p.n) gi = p.n - 1;            // clamp (keeps EXEC full)
    const float x1v = p.x1[gi];
    const float x2v = p.x2[gi];

    // Layer 1 (K=2): straight into D layout. Padded features (>=10) are exactly 0.
    float D[8];
#pragma unroll
    for (int r = 0; r < 8; ++r)
      D[r] = TANHF(__builtin_fmaf(x1v, w1s[r], __builtin_fmaf(x2v, w2s[r], b1s[r])));

    // 6 hidden layers: D(prev) -> B via 4 half-swaps, then 3x WMMA f32 16x16x4
    // with the bias pre-loaded as the accumulator (C) of the first WMMA.
#pragma unroll
    for (int L = 0; L < 6; ++L) {
      float t0 = D[0], t2 = D[2];  half_swap(t0, t2, loH);  // t0={f0,f2}  t2={f8,0}
      float t1 = D[1], t3 = D[3];  half_swap(t1, t3, loH);  // t1={f1,f3}  t3={f9,0}
      float t4 = D[4], t6 = D[6];  half_swap(t4, t6, loH);  // t4={f4,f6}
      float t5 = D[5], t7 = D[7];  half_swap(t5, t7, loH);  // t5={f5,f7}
      v2f b0; b0.x = t0; b0.y = t1;     // K = 0,2 / 1,3
      v2f b1; b1.x = t4; b1.y = t5;     // K = 4,6 / 5,7
      v2f b2; b2.x = t2; b2.y = t3;     // K = 8,10 / 9,11 (10,11 are zero)

      v8f c = abv[L];                   // bias as C-matrix: no post-add needed
      c = __builtin_amdgcn_wmma_f32_16x16x4_f32(false, A[L][0], false, b0,
                                                (short)0, c, false, false);
      c = __builtin_amdgcn_wmma_f32_16x16x4_f32(false, A[L][1], false, b1,
                                                (short)0, c, false, false);
      c = __builtin_amdgcn_wmma_f32_16x16x4_f32(false, A[L][2], false, b2,
                                                (short)0, c, false, false);
#pragma unroll
      for (int r = 0; r < 8; ++r) D[r] = TANHF(c[r]);
    }

    // Output layer: per-lane partial dot over this half's 8 features, fold halves.
    float acc = 0.0f;
#pragma unroll
    for (int r = 0; r < 8; ++r) acc = __builtin_fmaf(D[r], w9s[r], acc);
    float ra = acc, rb = acc;
    half_swap(ra, rb, loH);            // ra = {acc.lo, acc.lo}, rb = {acc.hi, acc.hi}
    const float tot = ra + rb + b9v;   // full 10-feature dot on every lane
    if (loH && (base + lane) < p.n) p.out[base + lane] = tot;
  }
}

extern "C" void kernel_launch(void* const* d_in, const int* in_sizes, int n_in,
                              void* d_out, int out_size, void* d_ws, size_t ws_size,
                              hipStream_t stream) {
  MlpParams p;
  p.x1    = (const float*)d_in[0];
  p.x2    = (const float*)d_in[1];
  p.w1    = (const float*)d_in[2];
  p.w2    = (const float*)d_in[3];
  p.b1    = (const float*)d_in[4];
  p.W3    = (const float*)d_in[5];   p.B3 = (const float*)d_in[6];
  p.W4    = (const float*)d_in[7];   p.B4 = (const float*)d_in[8];
  p.W5    = (const float*)d_in[9];   p.B5 = (const float*)d_in[10];
  p.W6    = (const float*)d_in[11];  p.B6 = (const float*)d_in[12];
  p.W7    = (const float*)d_in[13];  p.B7 = (const float*)d_in[14];
  p.W8    = (const float*)d_in[15];  p.B8 = (const float*)d_in[16];
  p.alpha = (const float*)d_in[17];
  p.w9    = (const float*)d_in[18];
  p.b9    = (const float*)d_in[19];
  p.out   = (float*)d_out;
  p.n     = in_sizes[0];

  const int blocks = 2048;   // 16384 wave32s -> ~15 tiles each of 250k tiles
  mlp_wmma_kernel<<<dim3(blocks), dim3(256), 0, stream>>>(p);
}